// SkeletonAttentionModule_82849919140577
// MI455X (gfx1250) — compile-verified
//
#include <hip/hip_runtime.h>
#include <cstdint>
#include <cstddef>

// Problem constants (fixed by the reference setup)
#define NB 64
#define CC 256
#define HH 64
#define WW 64
#define TT 64
#define VV 25
#define MM 2

// skeleton strides in floats: shape (N, 3, T, V, M)
#define SK_N (3 * TT * VV * MM) /* 9600 */
#define SK_C (TT * VV * MM)     /* 3200 */
#define SK_T (VV * MM)          /* 50   */

#if defined(__has_builtin)
#if __has_builtin(__builtin_amdgcn_global_load_async_to_lds_b128)
#define USE_ASYNC 1
#endif
#endif
#ifndef USE_ASYNC
#define USE_ASYNC 0
#endif

// 16-byte int vector type matching the builtin's expected pointee type
typedef int v4i_t __attribute__((vector_size(16)));
typedef __attribute__((address_space(1))) v4i_t* gv4_t;  // global (AS1)
typedef __attribute__((address_space(3))) v4i_t* lv4_t;  // LDS (AS3)

// s_wait_asynccnt via inline asm: ISA wait + compiler memory barrier so LDS
// reads are not hoisted above it. n is always a constant after unrolling.
__device__ __forceinline__ void wait_async_le(int n) {
  switch (n) {
    case 0: asm volatile("s_wait_asynccnt 0" ::: "memory"); break;
    case 1: asm volatile("s_wait_asynccnt 1" ::: "memory"); break;
    case 2: asm volatile("s_wait_asynccnt 2" ::: "memory"); break;
    default: asm volatile("s_wait_asynccnt 3" ::: "memory"); break;
  }
}

__global__ __launch_bounds__(256) void skel_attn_fused(
    const float* __restrict__ rgb, const float* __restrict__ skel,
    float* __restrict__ out) {
  const int tid = threadIdx.x;
  const int n = blockIdx.x >> 6;  // blockIdx.x / HH
  const int h = blockIdx.x & 63;  // blockIdx.x % HH

  __shared__ float s_xy[2];
  __shared__ __align__(16) float s_mask[WW];
#if USE_ASYNC
  __shared__ __align__(16) float4 s_buf[4][256];  // 16 KB, 4-slot ring
#endif

  // ---- Stage 1: wave 0 computes argmax joint position for batch n ----
  if (tid < 32) {
    const float* sk = skel + (size_t)n * SK_N;
    float d = -3.0e38f;
    int bv = tid;
    if (tid < VV) {
      float acc0 = 0.f, acc1 = 0.f;
#pragma unroll
      for (int c = 0; c < 3; ++c) {
        const float* f0 = sk + c * SK_C + tid * MM;                   // t = 0
        const float* fm = sk + c * SK_C + (TT / 2) * SK_T + tid * MM; // t = 32
        float d0 = fm[0] - f0[0];
        float d1 = fm[1] - f0[1];
        acc0 += d0 * d0;
        acc1 += d1 * d1;
      }
      d = 0.5f * (sqrtf(acc0) + sqrtf(acc1));  // mean over M of the two norms
    }
    // wave32 argmax reduction, first-index tie-break (matches jnp.argmax)
#pragma unroll
    for (int off = 16; off > 0; off >>= 1) {
      float od = __shfl_down(d, off, 32);
      int ov = __shfl_down(bv, off, 32);
      if (od > d || (od == d && ov < bv)) {
        d = od;
        bv = ov;
      }
    }
    if (tid == 0) {
      float jx = sk[0 * SK_C + (TT / 2) * SK_T + bv * MM + 0];  // body m=0
      float jy = sk[1 * SK_C + (TT / 2) * SK_T + bv * MM + 0];
      s_xy[0] = truncf(jx * (float)WW);
      s_xy[1] = truncf(jy * (float)HH);
    }
  }
  __syncthreads();

  // ---- Stage 2: build the 64-wide mask row for this h ----
  if (tid < WW) {
    const float inv2s2 = 1.0f / 184.32f;  // 1 / (2 * (0.15*64)^2)
    float dx = (float)tid - s_xy[0];
    float dy = (float)h - s_xy[1];
    float g = expf(-(dx * dx + dy * dy) * inv2s2);
    s_mask[tid] = 1.0f / (1.0f + expf(-g));  // sigmoid(gaussian)
  }
  __syncthreads();

  // ---- Stage 3: stream 256 channel-rows of (n,h), multiply by mask ----
  const int w4 = tid & 15;  // which float4 along W
  const int cr = tid >> 4;  // channel-row within a 16-row chunk
  const float4 m4 = *(const float4*)&s_mask[w4 * 4];

  const size_t base = (size_t)n * (size_t)(CC * HH * WW) + (size_t)h * WW;
  const float* gin = rgb + base;
  float* gout = out + base;

#if USE_ASYNC
  // 4-deep async DMA pipeline: each thread stages its own 16 B into LDS.
  // A lane only reads back bytes it staged itself, so no block barriers are
  // needed — just the per-wave async counter wait. The ds_load->store VGPR
  // dependency (compiler-inserted dscnt wait) guarantees slot reuse is safe.
  auto stage = [&](int i, int s) {
    size_t off = (size_t)(i * 16 + cr) * (HH * WW) + (size_t)(w4 * 4);
    gv4_t gp = (gv4_t)(uintptr_t)(gin + off);  // flat VA == global VA
    lv4_t lp = (lv4_t)(void*)&s_buf[s][tid];   // addrspacecast to LDS
    __builtin_amdgcn_global_load_async_to_lds_b128(gp, lp, 0, 0);
  };
  stage(0, 0);
  stage(1, 1);
  stage(2, 2);
#pragma unroll
  for (int i = 0; i < 16; ++i) {
    if (i + 3 < 16) stage(i + 3, (i + 3) & 3);
    // outstanding after issue = min(4, 16-i); chunk i done when cnt <= that-1
    wait_async_le((15 - i) < 3 ? (15 - i) : 3);
    float4 v = *(const float4*)&s_buf[i & 3][tid];
    size_t off = (size_t)(i * 16 + cr) * (HH * WW) + (size_t)(w4 * 4);
    float4 r;
    r.x = v.x * m4.x;
    r.y = v.y * m4.y;
    r.z = v.z * m4.z;
    r.w = v.w * m4.w;
    *(float4*)(gout + off) = r;
  }
#else
  // Fallback: straight b128 streaming loop
#pragma unroll 4
  for (int i = 0; i < 16; ++i) {
    size_t off = (size_t)(i * 16 + cr) * (HH * WW) + (size_t)(w4 * 4);
    float4 v = *(const float4*)(gin + off);
    float4 r;
    r.x = v.x * m4.x;
    r.y = v.y * m4.y;
    r.z = v.z * m4.z;
    r.w = v.w * m4.w;
    *(float4*)(gout + off) = r;
  }
#endif
}

extern "C" void kernel_launch(void* const* d_in, const int* in_sizes, int n_in,
                              void* d_out, int out_size, void* d_ws,
                              size_t ws_size, hipStream_t stream) {
  (void)in_sizes;
  (void)n_in;
  (void)out_size;
  (void)d_ws;
  (void)ws_size;
  const float* rgb = (const float*)d_in[0];
  const float* skel = (const float*)d_in[1];
  float* out = (float*)d_out;

  dim3 grid(NB * HH);  // one block per (n, h) row: 4096 blocks
  dim3 block(256);     // 8 wave32 waves
  skel_attn_fused<<<grid, block, 0, stream>>>(rgb, skel, out);
}